// HybridGNN_64338610094392
// MI455X (gfx1250) — compile-verified
//
#include <hip/hip_runtime.h>
#include <hip/hip_bf16.h>
#include <math.h>

typedef __attribute__((ext_vector_type(2))) float v2f;
typedef __attribute__((ext_vector_type(8))) float v8f;

#define NTOT_ 167936LL
#define ETOT_ 655360LL
#define BL_   8192      // B*L rows of RNN path
#define GDIM_ 384       // 3*HH

// ---------------------------------------------------------------------------
// fp32 WMMA GEMM:  Out = (ACCUM ? Out : 0) + alpha*(A@B) + bias, opt RELU.
//   A: M x K row-major.  B: TRANSB ? (N x K row-major, i.e. X@W^T) : (K x N).
//   M mult of 16, N mult of 32, K mult of 4.  One 16x32 tile per wave32
//   (two accumulators share one A fragment -> 2 v_wmma per k-step).
// Fragment layouts per CDNA5 ISA 7.12.2 (V_WMMA_F32_16X16X4_F32):
//   half = lane>>4, l = lane&15
//   A frag: a[j] = A[mt*16+l][k0 + 2*half + j]
//   B frag: b[j] = B[k0 + 2*half + j][col]
//   C/D:    c[v] = C[mt*16 + v + 8*half][col]
// ---------------------------------------------------------------------------
template <int TRANSB, int ACCUM, int RELU>
__global__ void gemm_wmma_f32(const float* __restrict__ A,
                              const float* __restrict__ Bm,
                              const float* __restrict__ bias,
                              float* __restrict__ Out,
                              int M, int K, int N, float alpha)
{
  const int wave = threadIdx.x >> 5;
  const int lane = threadIdx.x & 31;
  const int half = lane >> 4;
  const int l    = lane & 15;
  const int ntiles = N >> 5;                  // 32-wide column tiles
  const int tiles  = (M >> 4) * ntiles;
  const int tile   = blockIdx.x * 8 + wave;
  if (tile >= tiles) return;                  // wave-uniform exit (EXEC all-1 for WMMA)
  const int mt = tile / ntiles;
  const int nt = tile - mt * ntiles;
  const int row  = mt * 16 + l;
  const int col0 = nt * 32 + l;
  const int col1 = col0 + 16;

  v8f acc0 = {0.f, 0.f, 0.f, 0.f, 0.f, 0.f, 0.f, 0.f};
  v8f acc1 = acc0;

  const float* Ap = A + (long long)row * K + 2 * half;  // advance by k0
  const float* Bp0;
  const float* Bp1;
  if (TRANSB) {
    Bp0 = Bm + (long long)col0 * K + 2 * half;          // advance by k0
    Bp1 = Bp0 + 16LL * K;
  } else {
    Bp0 = Bm + (long long)(2 * half) * N + col0;        // advance by 4*N per step
    Bp1 = Bp0 + 16;
  }

#pragma unroll 4
  for (int k0 = 0; k0 < K; k0 += 4) {
    const v2f a = *(const v2f*)(Ap + k0);
    v2f b0, b1;
    if (TRANSB) {
      b0 = *(const v2f*)(Bp0 + k0);
      b1 = *(const v2f*)(Bp1 + k0);
    } else {
      b0.x = Bp0[0];
      b0.y = Bp0[N];
      b1.x = Bp1[0];
      b1.y = Bp1[N];
      Bp0 += 4LL * N;
      Bp1 += 4LL * N;
    }
    acc0 = __builtin_amdgcn_wmma_f32_16x16x4_f32(false, a, false, b0,
                                                 (short)0, acc0, false, false);
    acc1 = __builtin_amdgcn_wmma_f32_16x16x4_f32(false, a, false, b1,
                                                 (short)0, acc1, false, false);
  }

  const float bs0 = bias ? bias[col0] : 0.f;
  const float bs1 = bias ? bias[col1] : 0.f;
  float* Orow = Out + (long long)(mt * 16 + 8 * half) * N;
#pragma unroll
  for (int v = 0; v < 8; ++v) {
    float val0 = alpha * acc0[v] + bs0;
    float val1 = alpha * acc1[v] + bs1;
    float* o = Orow + (long long)v * N;
    if (ACCUM) { val0 += o[col0]; val1 += o[col1]; }
    if (RELU)  { val0 = fmaxf(val0, 0.f); val1 = fmaxf(val1, 0.f); }
    o[col0] = val0;
    o[col1] = val1;
  }
}

// ---------------------------------------------------------------------------
// Persistent BiGRU scan: one block per direction; 512 serial steps.
// gi: (2, 8192, 384) precomputed input gates.  whh: (2,384,128). bhh: (2,384).
// ys: (16,512,256): dir d writes columns [d*128, d*128+128).
// ---------------------------------------------------------------------------
__global__ void gru_scan(const float* __restrict__ gi,
                         const float* __restrict__ whh,
                         const float* __restrict__ bhh,
                         float* __restrict__ ys)
{
  const int d   = blockIdx.x;
  const int tid = threadIdx.x;                 // blockDim = 384
  const float* giD  = gi  + (long long)d * BL_ * GDIM_;
  const float* whhD = whh + d * GDIM_ * 128;
  const float* bhhD = bhh + d * GDIM_;

  __shared__ float h[2][16 * 128];
  __shared__ float gh[16 * GDIM_];

  for (int i = tid; i < 16 * 128; i += blockDim.x) h[0][i] = 0.f;
  __syncthreads();

  int p = 0;
  for (int t = 0; t < 512; ++t) {
    const int tt = d ? (511 - t) : t;
    // gh[b,g] = bhh[g] + h[b,:] . whh[g,:]
    for (int i = tid; i < 16 * GDIM_; i += blockDim.x) {
      const int b = i / GDIM_;
      const int g = i - b * GDIM_;
      const float* w  = whhD + g * 128;
      const float* hb = &h[p][b * 128];
      float s = bhhD[g];
#pragma unroll 4
      for (int k = 0; k < 128; ++k) s += hb[k] * w[k];
      gh[i] = s;
    }
    __syncthreads();
    for (int i = tid; i < 16 * 128; i += blockDim.x) {
      const int b = i >> 7;
      const int j = i & 127;
      const float* gib = giD + (long long)(b * 512 + tt) * GDIM_;
      const float ir  = gib[j];
      const float iz  = gib[128 + j];
      const float in_ = gib[256 + j];
      const float hr = gh[b * GDIM_ + j];
      const float hz = gh[b * GDIM_ + 128 + j];
      const float hn = gh[b * GDIM_ + 256 + j];
      const float r = 1.f / (1.f + expf(-(ir + hr)));
      const float z = 1.f / (1.f + expf(-(iz + hz)));
      const float n = tanhf(in_ + r * hn);
      const float hv = (1.f - z) * n + z * h[p][i];
      h[1 - p][i] = hv;
      ys[(long long)(b * 512 + tt) * 256 + d * 128 + j] = hv;
    }
    __syncthreads();
    p ^= 1;
  }
}

// ---------------------------------------------------------------------------
// Row LayerNorm over D=256: out = LN(relu?(in*preScale)) * w + b.  1 row/block.
// ---------------------------------------------------------------------------
__global__ void ln_rows(const float* __restrict__ in, const float* __restrict__ w,
                        const float* __restrict__ b, float* __restrict__ out,
                        float preScale, int doRelu)
{
  const int row = blockIdx.x;
  const int tid = threadIdx.x;                 // 256 threads
  float v = in[(long long)row * 256 + tid] * preScale;
  if (doRelu) v = fmaxf(v, 0.f);
  __shared__ float s1[256];
  __shared__ float s2[256];
  s1[tid] = v; s2[tid] = v * v;
  __syncthreads();
  for (int o = 128; o > 0; o >>= 1) {
    if (tid < o) { s1[tid] += s1[tid + o]; s2[tid] += s2[tid + o]; }
    __syncthreads();
  }
  const float mean = s1[0] * (1.f / 256.f);
  const float var  = s2[0] * (1.f / 256.f) - mean * mean;
  const float inv  = rsqrtf(var + 1e-5f);
  out[(long long)row * 256 + tid] = (v - mean) * inv * w[tid] + b[tid];
}

// accum[row,:] += in[row,:] / max(||in[row,:]||2, 1e-12)   (D = 256)
__global__ void l2norm_add(const float* __restrict__ in, float* __restrict__ accum)
{
  const int row = blockIdx.x;
  const int tid = threadIdx.x;
  const float v = in[(long long)row * 256 + tid];
  __shared__ float s[256];
  s[tid] = v * v;
  __syncthreads();
  for (int o = 128; o > 0; o >>= 1) {
    if (tid < o) s[tid] += s[tid + o];
    __syncthreads();
  }
  const float denom = fmaxf(sqrtf(s[0]), 1e-12f);
  accum[(long long)row * 256 + tid] += v / denom;
}

// Edge scatter: agg[dst,:] += feat[src,:]; cnt[dst] += 1 (once per edge).
__global__ void scatter_add(const float* __restrict__ feat,
                            const int* __restrict__ src,
                            const int* __restrict__ dst,
                            float* __restrict__ agg, float* __restrict__ cnt,
                            long long E, int D)
{
  const long long gid = (long long)blockIdx.x * blockDim.x + threadIdx.x;
  if (gid >= E * (long long)D) return;
  const long long e = gid / D;
  const int f = (int)(gid - e * D);
  const int s = src[e];
  const int d0 = dst[e];
  unsafeAtomicAdd(&agg[(long long)d0 * D + f], feat[(long long)s * D + f]);
  if (f == 0) unsafeAtomicAdd(&cnt[d0], 1.0f);
}

__global__ void mean_div(float* __restrict__ agg, const float* __restrict__ cnt,
                         long long total, int D)
{
  const long long gid = (long long)blockIdx.x * blockDim.x + threadIdx.x;
  if (gid >= total) return;
  const long long r = gid / D;
  agg[gid] /= fmaxf(cnt[r], 1.f);
}

__global__ void fill_zero(float* __restrict__ p, long long n)
{
  const long long gid = (long long)blockIdx.x * blockDim.x + threadIdx.x;
  if (gid < n) p[gid] = 0.f;
}

// ---------------------------------------------------------------------------
// Host-side orchestration
// ---------------------------------------------------------------------------
static inline void launch_gemm(const float* A, const float* Bm, const float* bias,
                               float* Out, int M, int K, int N,
                               int transB, int accum, int relu, float alpha,
                               hipStream_t s)
{
  const int tiles  = (M / 16) * (N / 32);
  const int blocks = (tiles + 7) / 8;
  if (transB)
    gemm_wmma_f32<1, 0, 0><<<blocks, 256, 0, s>>>(A, Bm, bias, Out, M, K, N, alpha);
  else if (accum)
    gemm_wmma_f32<0, 1, 0><<<blocks, 256, 0, s>>>(A, Bm, bias, Out, M, K, N, alpha);
  else if (relu)
    gemm_wmma_f32<0, 0, 1><<<blocks, 256, 0, s>>>(A, Bm, bias, Out, M, K, N, alpha);
  else
    gemm_wmma_f32<0, 0, 0><<<blocks, 256, 0, s>>>(A, Bm, bias, Out, M, K, N, alpha);
}

static inline void launch_zero(float* p, long long n, hipStream_t s)
{
  fill_zero<<<(unsigned)((n + 255) / 256), 256, 0, s>>>(p, n);
}

extern "C" void kernel_launch(void* const* d_in, const int* in_sizes, int n_in,
                              void* d_out, int out_size, void* d_ws, size_t ws_size,
                              hipStream_t stream)
{
  (void)in_sizes; (void)n_in; (void)out_size; (void)ws_size;

  const float* x    = (const float*)d_in[0];    // (NTOT, 64)
  const int*   ei   = (const int*)  d_in[1];    // (3, 2, ETOT)
  const float* s0pw = (const float*)d_in[4];    // (3,64,64)
  const float* s0pb = (const float*)d_in[5];    // (3,64)
  const float* s0lw = (const float*)d_in[6];    // (3,64,256)
  const float* s0lb = (const float*)d_in[7];    // (3,256)
  const float* s0rw = (const float*)d_in[8];    // (3,64,256)
  const float* slw  = (const float*)d_in[9];    // (2,3,256,256)
  const float* slb  = (const float*)d_in[10];   // (2,3,256)
  const float* srw  = (const float*)d_in[11];   // (2,3,256,256)
  const float* lnw  = (const float*)d_in[12];   // (2,256)
  const float* lnb  = (const float*)d_in[13];   // (2,256)
  const float* wih0 = (const float*)d_in[14];   // (2,384,64)
  const float* whh0 = (const float*)d_in[15];   // (2,384,128)
  const float* bih0 = (const float*)d_in[16];   // (2,384)
  const float* bhh0 = (const float*)d_in[17];   // (2,384)
  const float* wih1 = (const float*)d_in[18];   // (2,384,256)
  const float* whh1 = (const float*)d_in[19];   // (2,384,128)
  const float* bih1 = (const float*)d_in[20];   // (2,384)
  const float* bhh1 = (const float*)d_in[21];   // (2,384)
  const float* rnw  = (const float*)d_in[22];
  const float* rnb  = (const float*)d_in[23];
  const float* m1w  = (const float*)d_in[24];
  const float* m1b  = (const float*)d_in[25];
  const float* mlnw = (const float*)d_in[26];
  const float* mlnb = (const float*)d_in[27];
  const float* m2w  = (const float*)d_in[28];
  const float* m2b  = (const float*)d_in[29];
  const float* cw   = (const float*)d_in[30];   // (512,256)
  const float* cb   = (const float*)d_in[31];   // (256)
  float* out = (float*)d_out;                   // (8192,256)
  float* ws  = (float*)d_ws;

  // ---- workspace layout (floats) ----
  float* gi   = ws;                       // 2*8192*384
  float* ys0  = gi   + 2LL * BL_ * GDIM_; // 8192*256
  float* ys1  = ys0  + 2097152;
  float* bufA = ys1  + 2097152;           // LN / t2 scratch
  float* bufB = bufA + 2097152;           // mlp1 scratch
  float* xrnn = bufB + 2097152;
  float* xp   = xrnn + 2097152;           // NTOT*64
  float* agg  = xp   + NTOT_ * 64;        // max(NTOT*64, 106496*256) = 27262976
  float* cnt  = agg  + 27262976LL;        // NTOT
  float* tmpF = cnt  + NTOT_;             // NTOT*256 (layer-0 tmp, then featBuf)
  float* accB = tmpF + NTOT_ * 256;       // NTOT*256 (accum0, then accumL)

  // =================== RNN path ===================
  for (int d = 0; d < 2; ++d) {
    launch_gemm(x, wih0 + d * GDIM_ * 64, bih0 + d * GDIM_,
                gi + (long long)d * BL_ * GDIM_, BL_, 64, GDIM_, 1, 0, 0, 1.f, stream);
  }
  gru_scan<<<2, 384, 0, stream>>>(gi, whh0, bhh0, ys0);
  for (int d = 0; d < 2; ++d) {
    launch_gemm(ys0, wih1 + d * GDIM_ * 256, bih1 + d * GDIM_,
                gi + (long long)d * BL_ * GDIM_, BL_, 256, GDIM_, 1, 0, 0, 1.f, stream);
  }
  gru_scan<<<2, 384, 0, stream>>>(gi, whh1, bhh1, ys1);

  ln_rows<<<BL_, 256, 0, stream>>>(ys1, rnw, rnb, bufA, 1.f, 0);
  launch_gemm(bufA, m1w, m1b, bufB, BL_, 256, 256, 0, 0, 1, 1.f, stream);
  ln_rows<<<BL_, 256, 0, stream>>>(bufB, mlnw, mlnb, bufA, 1.f, 0);
  launch_gemm(bufA, m2w, m2b, xrnn, BL_, 256, 256, 0, 0, 0, 1.f, stream);

  // =================== GNN layer 0 (SAGE with proj + l2) ===================
  launch_zero(accB, NTOT_ * 256, stream);
  for (int t = 0; t < 3; ++t) {
    const int* srcp = ei + (long long)t * 2 * ETOT_;
    const int* dstp = srcp + ETOT_;
    launch_gemm(x, s0pw + t * 64 * 64, s0pb + t * 64, xp,
                (int)NTOT_, 64, 64, 0, 0, 1, 1.f, stream);
    launch_zero(agg, NTOT_ * 64, stream);
    launch_zero(cnt, NTOT_, stream);
    {
      const long long work = ETOT_ * 64;
      scatter_add<<<(unsigned)((work + 255) / 256), 256, 0, stream>>>(
          xp, srcp, dstp, agg, cnt, ETOT_, 64);
    }
    mean_div<<<(unsigned)((NTOT_ * 64 + 255) / 256), 256, 0, stream>>>(
        agg, cnt, NTOT_ * 64, 64);
    launch_gemm(agg, s0lw + t * 64 * 256, s0lb + t * 256, tmpF,
                (int)NTOT_, 64, 256, 0, 0, 0, 1.f, stream);
    launch_gemm(x, s0rw + t * 64 * 256, nullptr, tmpF,
                (int)NTOT_, 64, 256, 0, 1, 0, 1.f, stream);
    l2norm_add<<<(unsigned)NTOT_, 256, 0, stream>>>(tmpF, accB);
  }
  // feat = LN(relu(accum/3))  -> featBuf (= tmpF)
  ln_rows<<<(unsigned)NTOT_, 256, 0, stream>>>(accB, lnw, lnb, tmpF, 1.f / 3.f, 1);

  // =================== GNN layers 1,2 ===================
  const long long nl[2] = {106496, 40960};
  const long long el[2] = {393216, 131072};
  for (int i = 0; i < 2; ++i) {
    launch_zero(accB, nl[i] * 256, stream);
    for (int t = 0; t < 3; ++t) {
      const int* srcp = ei + (long long)t * 2 * ETOT_;
      const int* dstp = srcp + ETOT_;
      launch_zero(agg, nl[i] * 256, stream);
      launch_zero(cnt, nl[i], stream);
      const long long work = el[i] * 256;
      scatter_add<<<(unsigned)((work + 255) / 256), 256, 0, stream>>>(
          tmpF, srcp, dstp, agg, cnt, el[i], 256);
      mean_div<<<(unsigned)((nl[i] * 256 + 255) / 256), 256, 0, stream>>>(
          agg, cnt, nl[i] * 256, 256);
      launch_gemm(agg, slw + ((long long)i * 3 + t) * 256 * 256,
                  slb + ((long long)i * 3 + t) * 256, accB,
                  (int)nl[i], 256, 256, 0, 1, 0, 1.f, stream);
      launch_gemm(tmpF, srw + ((long long)i * 3 + t) * 256 * 256, nullptr, accB,
                  (int)nl[i], 256, 256, 0, 1, 0, 1.f, stream);
    }
    if (i == 0) {
      ln_rows<<<(unsigned)nl[i], 256, 0, stream>>>(accB, lnw + 256, lnb + 256,
                                                   tmpF, 1.f / 3.f, 1);
    }
  }

  // =================== concat head ===================
  // out = x_rnn @ cat_w[0:256] + cat_b  +  (accumL/3) @ cat_w[256:512]
  launch_gemm(xrnn, cw, cb, out, BL_, 256, 256, 0, 0, 0, 1.f, stream);
  launch_gemm(accB, cw + 256 * 256, nullptr, out, BL_, 256, 256, 0, 1, 0,
              1.f / 3.f, stream);
}